// FraudGNN_36825049596435
// MI455X (gfx1250) — compile-verified
//
#include <hip/hip_runtime.h>
#include <hip/hip_bf16.h>

#define IN_F 128
#define HID  128
#define N_CLS 2

typedef __attribute__((ext_vector_type(2))) float v2f;
typedef __attribute__((ext_vector_type(8))) float v8f;

// ---------------------------------------------------------------- degree ----
__global__ void gcn_zero_f32(float* __restrict__ p, int n) {
    int i = blockIdx.x * blockDim.x + threadIdx.x;
    if (i < n) p[i] = 0.0f;
}

__global__ void gcn_deg_accum(const int* __restrict__ dst, float* __restrict__ deg, int nE) {
    int e = blockIdx.x * blockDim.x + threadIdx.x;
    if (e < nE) atomicAdd(&deg[dst[e]], 1.0f);
}

__global__ void gcn_dinv_finalize(float* __restrict__ dinv, int n) {
    int i = blockIdx.x * blockDim.x + threadIdx.x;
    if (i < n) dinv[i] = rsqrtf(dinv[i] + 1.0f);   // +1 = self loop; always > 0
}

// ------------------------------------------------------- GEMM1 via WMMA -----
// xw[100000,128] = x[100000,128] @ W1[128,128], f32 WMMA 16x16x4.
// One wave per 16-ROW strip, 8 column-tile accumulators -> A loaded once,
// 256 wmma per wave. 100000 = 6250*16 exactly; 6250 waves / 2 per block.
__global__ __launch_bounds__(64) void gcn_gemm1_wmma(const float* __restrict__ x,
                                                     const float* __restrict__ W1,
                                                     float* __restrict__ xw) {
    const int wave = blockIdx.x * (blockDim.x >> 5) + (threadIdx.x >> 5);  // row tile
    const int lane = threadIdx.x & 31;
    const int half = lane >> 4;            // 0: K pair {0,1}, 1: K pair {2,3}
    const int lm   = lane & 15;
    const int row0 = wave * 16;

    const float* __restrict__ xrow = x + (size_t)(row0 + lm) * IN_F;

    v8f acc[8];
#pragma unroll
    for (int cb = 0; cb < 8; ++cb) acc[cb] = (v8f){};

#pragma unroll 4
    for (int k0 = 0; k0 < IN_F; k0 += 4) {
        const int ka = k0 + 2 * half;
        // A 16x4: lane lm holds row M=lm, floats K=ka, ka+1 (one b64 load)
        v2f a = *(const v2f*)(xrow + ka);
        const float* __restrict__ w0 = W1 + (size_t)(ka + 0) * HID + lm;
        const float* __restrict__ w1 = W1 + (size_t)(ka + 1) * HID + lm;
#pragma unroll
        for (int cb = 0; cb < 8; ++cb) {
            // B 4x16: lane lm holds col N=cb*16+lm, rows K=ka, ka+1
            v2f b;
            b.x = w0[cb * 16];
            b.y = w1[cb * 16];
            acc[cb] = __builtin_amdgcn_wmma_f32_16x16x4_f32(false, a, false, b,
                                                            (short)0, acc[cb], false, false);
        }
    }
    // C/D: VGPR r -> M = r (lanes 0-15) / r+8 (lanes 16-31), N = lm
#pragma unroll
    for (int cb = 0; cb < 8; ++cb) {
#pragma unroll
        for (int r = 0; r < 8; ++r) {
            xw[(size_t)(row0 + r + 8 * half) * HID + cb * 16 + lm] = acc[cb][r];
        }
    }
}

// ------------------------------------------------- layer-1 aggregation ------
// Self-loop term initializes h (non-atomic full write, also clears buffer).
__global__ void gcn_selfloop_h(const float* __restrict__ xw, const float* __restrict__ dinv,
                               float* __restrict__ h, int n) {
    int idx = blockIdx.x * blockDim.x + threadIdx.x;     // n * 32 threads
    int node = idx >> 5;
    if (node >= n) return;
    int lane = idx & 31;
    float s = dinv[node];
    s = s * s;
    const float4 v = *(const float4*)(xw + (size_t)node * HID + lane * 4);
    float4 o = make_float4(v.x * s, v.y * s, v.z * s, v.w * s);
    *(float4*)(h + (size_t)node * HID + lane * 4) = o;
}

// One wave per edge: coalesced 512B gather from xw (L2-resident), 4 f32 atomics/lane.
__global__ void gcn_agg1(const int* __restrict__ src, const int* __restrict__ dst,
                         const float* __restrict__ dinv, const float* __restrict__ xw,
                         float* __restrict__ h, int nE) {
    int e = (blockIdx.x * blockDim.x + threadIdx.x) >> 5;
    if (e >= nE) return;
    int lane = threadIdx.x & 31;
    int s = src[e], d = dst[e];
    float norm = dinv[s] * dinv[d];
    const float4 v = *(const float4*)(xw + (size_t)s * HID + lane * 4);
    float* o = h + (size_t)d * HID + lane * 4;
    atomicAdd(o + 0, v.x * norm);
    atomicAdd(o + 1, v.y * norm);
    atomicAdd(o + 2, v.z * norm);
    atomicAdd(o + 3, v.w * norm);
}

// ------------------------------------------- fused bias+ReLU+GEMM2 ----------
// zw[n,2] = relu(h + b1) @ W2[128,2]. One wave per node, shfl reduction.
__global__ void gcn_gemm2(const float* __restrict__ h, const float* __restrict__ b1,
                          const float* __restrict__ W2, float* __restrict__ zw, int n) {
    int node = (blockIdx.x * blockDim.x + threadIdx.x) >> 5;
    if (node >= n) return;
    int lane = threadIdx.x & 31;
    const float4 hv = *(const float4*)(h + (size_t)node * HID + lane * 4);
    const float4 bv = *(const float4*)(b1 + lane * 4);
    float f0 = fmaxf(hv.x + bv.x, 0.0f);
    float f1 = fmaxf(hv.y + bv.y, 0.0f);
    float f2 = fmaxf(hv.z + bv.z, 0.0f);
    float f3 = fmaxf(hv.w + bv.w, 0.0f);
    int k = lane * 4;
    float a0 = f0 * W2[(k + 0) * 2 + 0] + f1 * W2[(k + 1) * 2 + 0]
             + f2 * W2[(k + 2) * 2 + 0] + f3 * W2[(k + 3) * 2 + 0];
    float a1 = f0 * W2[(k + 0) * 2 + 1] + f1 * W2[(k + 1) * 2 + 1]
             + f2 * W2[(k + 2) * 2 + 1] + f3 * W2[(k + 3) * 2 + 1];
#pragma unroll
    for (int off = 16; off > 0; off >>= 1) {
        a0 += __shfl_down(a0, off, 32);
        a1 += __shfl_down(a1, off, 32);
    }
    if (lane == 0) {
        zw[(size_t)node * 2 + 0] = a0;
        zw[(size_t)node * 2 + 1] = a1;
    }
}

// ------------------------------------------------- layer-2 aggregation ------
__global__ void gcn_selfloop_out(const float* __restrict__ zw, const float* __restrict__ dinv,
                                 float* __restrict__ out, int n) {
    int i = blockIdx.x * blockDim.x + threadIdx.x;
    if (i >= n) return;
    float s = dinv[i];
    s = s * s;
    out[(size_t)i * 2 + 0] = zw[(size_t)i * 2 + 0] * s;
    out[(size_t)i * 2 + 1] = zw[(size_t)i * 2 + 1] * s;
}

__global__ void gcn_agg2(const int* __restrict__ src, const int* __restrict__ dst,
                         const float* __restrict__ dinv, const float* __restrict__ zw,
                         float* __restrict__ out, int nE) {
    int e = blockIdx.x * blockDim.x + threadIdx.x;
    if (e >= nE) return;
    int s = src[e], d = dst[e];
    float norm = dinv[s] * dinv[d];
    atomicAdd(out + (size_t)d * 2 + 0, zw[(size_t)s * 2 + 0] * norm);
    atomicAdd(out + (size_t)d * 2 + 1, zw[(size_t)s * 2 + 1] * norm);
}

// --------------------------------------------- +b2 and log_softmax ----------
__global__ void gcn_logsoftmax(float* __restrict__ out, const float* __restrict__ b2, int n) {
    int i = blockIdx.x * blockDim.x + threadIdx.x;
    if (i >= n) return;
    float l0 = out[(size_t)i * 2 + 0] + b2[0];
    float l1 = out[(size_t)i * 2 + 1] + b2[1];
    float m = fmaxf(l0, l1);
    float lse = m + logf(expf(l0 - m) + expf(l1 - m));
    out[(size_t)i * 2 + 0] = l0 - lse;
    out[(size_t)i * 2 + 1] = l1 - lse;
}

// ----------------------------------------------------------------------------
extern "C" void kernel_launch(void* const* d_in, const int* in_sizes, int n_in,
                              void* d_out, int out_size, void* d_ws, size_t ws_size,
                              hipStream_t stream) {
    const float* x  = (const float*)d_in[0];
    const float* W1 = (const float*)d_in[1];
    const float* b1 = (const float*)d_in[2];
    const float* W2 = (const float*)d_in[3];
    const float* b2 = (const float*)d_in[4];
    const int*   ei = (const int*)d_in[5];

    const int n  = in_sizes[0] / IN_F;       // 100000
    const int nE = in_sizes[5] / 2;          // 1600000
    const int* src = ei;
    const int* dst = ei + nE;

    // workspace layout (256B aligned)
    char* ws = (char*)d_ws;
    float* xw   = (float*)(ws);                                     // n*128 f32
    float* h    = (float*)(ws + (size_t)n * HID * 4);               // n*128 f32
    float* dinv = (float*)(ws + (size_t)2 * n * HID * 4);           // n f32
    float* zw   = (float*)(ws + (size_t)2 * n * HID * 4 + 512000);  // n*2 f32
    float* out  = (float*)d_out;                                    // n*2 f32

    const int B = 256;

    // 1) degrees -> dinv = rsqrt(deg+1)
    gcn_zero_f32<<<(n + B - 1) / B, B, 0, stream>>>(dinv, n);
    gcn_deg_accum<<<(nE + B - 1) / B, B, 0, stream>>>(dst, dinv, nE);
    gcn_dinv_finalize<<<(n + B - 1) / B, B, 0, stream>>>(dinv, n);

    // 2) xw = x @ W1  (WMMA f32; 6250 row-strip waves, 2 waves / 64-thread block)
    {
        int rowTiles = n / 16;               // 6250
        gcn_gemm1_wmma<<<rowTiles / 2, 64, 0, stream>>>(x, W1, xw);
    }

    // 3) h = self-loop init, then edge scatter
    gcn_selfloop_h<<<((size_t)n * 32 + B - 1) / B, B, 0, stream>>>(xw, dinv, h, n);
    gcn_agg1<<<((size_t)nE * 32 + B - 1) / B, B, 0, stream>>>(src, dst, dinv, xw, h, nE);

    // 4) zw = relu(h + b1) @ W2
    gcn_gemm2<<<((size_t)n * 32 + B - 1) / B, B, 0, stream>>>(h, b1, W2, zw, n);

    // 5) logits accumulated in d_out
    gcn_selfloop_out<<<(n + B - 1) / B, B, 0, stream>>>(zw, dinv, out, n);
    gcn_agg2<<<(nE + B - 1) / B, B, 0, stream>>>(src, dst, dinv, zw, out, nE);

    // 6) + b2, log_softmax in place
    gcn_logsoftmax<<<(n + B - 1) / B, B, 0, stream>>>(out, b2, n);
}